// NMSPostProcess_6863357739224
// MI455X (gfx1250) — compile-verified
//
#include <hip/hip_runtime.h>
#include <math.h>

// Problem constants (from reference)
#define BS   128
#define NQ   900
#define NC   91
#define NTOT (NQ * NC)       // 81900 per image
#define PRE  10000
#define POST 100
#define IOU_THRESH 0.7f

#define NTHREADS 1024        // 32 wave32 waves per workgroup
#define TILE     4096        // floats staged in LDS per histogram tile (16 KB)

typedef int v4i __attribute__((ext_vector_type(4)));

// ---------------------------------------------------------------------------
// CDNA5 helpers: async global->LDS staging + ASYNCcnt waits.
// Builtin signature (probed in round 1): (v4i AS1* gsrc, v4i AS3* lds,
// int offset, int cpol).
// ---------------------------------------------------------------------------
__device__ __forceinline__ void wait_async_le(int n) {  // n is compile-time 0/1
#if __has_builtin(__builtin_amdgcn_s_wait_asynccnt)
  if (n == 0) __builtin_amdgcn_s_wait_asynccnt(0);
  else        __builtin_amdgcn_s_wait_asynccnt(1);
#else
  if (n == 0) asm volatile("s_wait_asynccnt 0" ::: "memory");
  else        asm volatile("s_wait_asynccnt 1" ::: "memory");
#endif
}

__device__ __forceinline__ void async_copy16(const float* __restrict__ gsrc,
                                             float* __restrict__ ldst) {
#if __has_builtin(__builtin_amdgcn_global_load_async_to_lds_b128)
  __builtin_amdgcn_global_load_async_to_lds_b128(
      (__attribute__((address_space(1))) v4i*)gsrc,
      (__attribute__((address_space(3))) v4i*)ldst,
      /*offset=*/0, /*cpol=*/0);
#else
  *(float4*)ldst = *(const float4*)gsrc;   // fallback: synchronous copy
#endif
}

// Monotone unsigned key for float compare (handles negatives correctly)
__device__ __forceinline__ unsigned fkey(float f) {
  unsigned u = __float_as_uint(f);
  return (u & 0x80000000u) ? ~u : (u | 0x80000000u);
}

// ---------------------------------------------------------------------------
// One workgroup per image. Phases:
//  1) 4-round radix-select (8 bits/round) of the PRE-th largest logit key.
//     Tiles are staged global->LDS with DOUBLE-BUFFERED async copies: issue
//     tile t+1, s_wait_asynccnt<=1 retires tile t, histogram it, barrier.
//  2) Compaction of the top-PRE candidates: sigmoid score, label, scaled
//     xyxy box, area. Scores/labels/areas live in LDS; boxes in workspace.
//  3) 100 serial NMS rounds: block-wide argmax (wave32 shuffle + LDS tree),
//     emit pick, suppress same-label boxes with IoU > 0.7 (equivalent to the
//     reference's per-class offset trick: IoU/areas are translation
//     invariant, so cross-class IoU is identically zero).
//  No WMMA: this workload has no matmul structure; the CDNA5 features that
//  apply are the async data movers, split wait counters, and big LDS.
// ---------------------------------------------------------------------------
__global__ __launch_bounds__(NTHREADS)
void nms_postprocess_kernel(const float* __restrict__ logits,
                            const float* __restrict__ pboxes,
                            const float* __restrict__ tsizes,
                            float* __restrict__ out,
                            float* __restrict__ ws_boxes) {
  __shared__ float    s_sc[PRE];          // 40 KB  candidate scores (-inf = dead)
  __shared__ int      s_lb[PRE];          // 40 KB  candidate labels
  __shared__ float    s_ar[PRE];          // 40 KB  candidate areas
  __shared__ float    s_stage[2][TILE];   // 32 KB  double-buffered staging
  __shared__ unsigned s_hist[256];
  __shared__ float    s_redv[NTHREADS / 32];
  __shared__ int      s_redi[NTHREADS / 32];
  __shared__ unsigned s_prefix, s_remaining, s_cntG, s_cntE;
  __shared__ int      s_best;

  const int b = blockIdx.x;
  const int t = threadIdx.x;
  const float* lg = logits + (size_t)b * NTOT;

  // ---------------- Phase 1: radix-select threshold ----------------
  if (t == 0) { s_prefix = 0u; s_remaining = PRE; }
  __syncthreads();

  const int NFULL = NTOT / TILE;                 // 19 full tiles
  for (int r = 0; r < 4; ++r) {
    const int shift = 24 - 8 * r;
    if (t < 256) s_hist[t] = 0u;
    __syncthreads();
    const unsigned pref = s_prefix;

    // Pipeline prologue: stage tile 0 into buffer 0.
    async_copy16(lg + t * 4, &s_stage[0][t * 4]);

    for (int tile = 0; tile < NFULL; ++tile) {
      const int cur = tile & 1;
      if (tile + 1 < NFULL) {
        // Issue next tile into the other buffer before consuming this one.
        async_copy16(lg + (tile + 1) * TILE + t * 4, &s_stage[cur ^ 1][t * 4]);
        wait_async_le(1);      // retire tile `tile` only; tile+1 stays in flight
      } else {
        wait_async_le(0);      // last tile: drain
      }
      __syncthreads();         // all waves' slices of tile `tile` visible

      for (int j = t; j < TILE; j += NTHREADS) {
        unsigned k = fkey(s_stage[cur][j]);
        if ((unsigned)(((unsigned long long)k) >> (shift + 8)) == pref)
          atomicAdd(&s_hist[(k >> shift) & 255u], 1u);
      }
      __syncthreads();         // tile consumed; safe to re-issue at next iter
    }
    // Tail elements straight from global (L2 resident after round 0)
    for (int j = NFULL * TILE + t; j < NTOT; j += NTHREADS) {
      unsigned k = fkey(lg[j]);
      if ((unsigned)(((unsigned long long)k) >> (shift + 8)) == pref)
        atomicAdd(&s_hist[(k >> shift) & 255u], 1u);
    }
    __syncthreads();

    if (t == 0) {
      unsigned rem = s_remaining, cum = 0u;
      int sel = 0;
      for (int j2 = 255; j2 >= 0; --j2) {
        cum += s_hist[j2];
        if (cum >= rem) { sel = j2; s_remaining = rem - (cum - s_hist[j2]); break; }
      }
      s_prefix = (pref << 8) | (unsigned)sel;
    }
    __syncthreads();
  }

  const unsigned Tkey = s_prefix;          // key of the PRE-th largest logit
  const unsigned nEq  = s_remaining;       // how many ==Tkey to admit
  const unsigned nG   = PRE - nEq;         // count strictly greater

  // ---------------- Phase 2: compaction ----------------
  if (t == 0) { s_cntG = 0u; s_cntE = 0u; }
  __syncthreads();

  const float img_h = tsizes[b * 2 + 0];
  const float img_w = tsizes[b * 2 + 1];
  float4* wbx = (float4*)ws_boxes + (size_t)b * PRE;

  for (int j = t; j < NTOT; j += NTHREADS) {
    __builtin_prefetch(&lg[j + 8 * NTHREADS], 0, 1);   // global_prefetch_b8
    const float x = lg[j];
    const unsigned k = fkey(x);
    int slot = -1;
    if (k > Tkey) {
      slot = (int)atomicAdd(&s_cntG, 1u);
    } else if (k == Tkey) {
      unsigned e = atomicAdd(&s_cntE, 1u);
      if (e < nEq) slot = (int)(nG + e);
    }
    if (slot >= 0) {
      const int q = j / NC;
      const int c = j - q * NC;
      const float* bp = pboxes + ((size_t)b * NQ + q) * 4;
      const float cx = bp[0], cy = bp[1], w = bp[2], h = bp[3];
      const float x1 = (cx - 0.5f * w) * img_w;
      const float y1 = (cy - 0.5f * h) * img_h;
      const float x2 = (cx + 0.5f * w) * img_w;
      const float y2 = (cy + 0.5f * h) * img_h;
      s_sc[slot] = 1.0f / (1.0f + __expf(-x));
      s_lb[slot] = c;
      s_ar[slot] = (x2 - x1) * (y2 - y1);
      wbx[slot]  = make_float4(x1, y1, x2, y2);
    }
  }
  __syncthreads();

  // ---------------- Phase 3: greedy NMS, POST picks ----------------
  const int wave = t >> 5;
  const int lane = t & 31;
  const float NEG_INF = -__builtin_inff();

  for (int k = 0; k < POST; ++k) {
    // block argmax over s_sc
    float bv = NEG_INF; int bi = 0;
    for (int j = t; j < PRE; j += NTHREADS) {
      float v = s_sc[j];
      if (v > bv) { bv = v; bi = j; }
    }
    for (int off = 16; off > 0; off >>= 1) {       // wave32 reduction
      float ov = __shfl_down(bv, off, 32);
      int   oi = __shfl_down(bi, off, 32);
      if (ov > bv) { bv = ov; bi = oi; }
    }
    if (lane == 0) { s_redv[wave] = bv; s_redi[wave] = bi; }
    __syncthreads();
    if (t == 0) {
      float mv = s_redv[0]; int mi = s_redi[0];
      for (int wv = 1; wv < NTHREADS / 32; ++wv)
        if (s_redv[wv] > mv) { mv = s_redv[wv]; mi = s_redi[wv]; }
      s_best = mi;
      const float4 bb = wbx[mi];
      out[(size_t)b * POST + k]                          = mv;              // scores
      out[(size_t)BS * POST + (size_t)b * POST + k]      = (float)s_lb[mi]; // labels
      float* ob = out + (size_t)2 * BS * POST + ((size_t)b * POST + k) * 4; // boxes
      ob[0] = bb.x; ob[1] = bb.y; ob[2] = bb.z; ob[3] = bb.w;
    }
    __syncthreads();

    const int    best = s_best;
    const float4 bb   = wbx[best];
    const int    bl   = s_lb[best];
    const float  ba   = s_ar[best];
    for (int j = t; j < PRE; j += NTHREADS) {
      if (s_lb[j] == bl) {                 // cross-class IoU is forced 0 in ref
        const float4 oj = wbx[j];
        const float xx1 = fmaxf(bb.x, oj.x), yy1 = fmaxf(bb.y, oj.y);
        const float xx2 = fminf(bb.z, oj.z), yy2 = fminf(bb.w, oj.w);
        const float inter = fmaxf(xx2 - xx1, 0.0f) * fmaxf(yy2 - yy1, 0.0f);
        const float uni   = ba + s_ar[j] - inter;
        const float iou   = inter / fmaxf(uni, 1e-9f);
        if (iou > IOU_THRESH) s_sc[j] = NEG_INF;   // best suppresses itself too
      }
    }
    __syncthreads();
  }
}

// ---------------------------------------------------------------------------
extern "C" void kernel_launch(void* const* d_in, const int* in_sizes, int n_in,
                              void* d_out, int out_size, void* d_ws, size_t ws_size,
                              hipStream_t stream) {
  (void)in_sizes; (void)n_in; (void)out_size; (void)ws_size;
  const float* logits = (const float*)d_in[0];   // [BS, NQ, NC]
  const float* pboxes = (const float*)d_in[1];   // [BS, NQ, 4] cxcywh
  const float* tsz    = (const float*)d_in[2];   // [BS, 2] (h, w)
  float* out = (float*)d_out;                    // scores | labels | boxes
  float* ws  = (float*)d_ws;                     // BS*PRE float4 candidate boxes

  nms_postprocess_kernel<<<dim3(BS), dim3(NTHREADS), 0, stream>>>(
      logits, pboxes, tsz, out, ws);
}